// GaussianSample_20272245637273
// MI455X (gfx1250) — compile-verified
//
#include <hip/hip_runtime.h>
#include <hip/hip_bf16.h>

#define N_NODES 100000
#define N_EDGES 3200000
#define D_IN    256
#define D_OUT   64

typedef __attribute__((ext_vector_type(2))) float v2f;
typedef __attribute__((ext_vector_type(4))) float v4f;
typedef __attribute__((ext_vector_type(8))) float v8f;

// ---------------------------------------------------------------------------
// Degree: deg[i] = 1 (self loop) + #edges with dst == i
// ---------------------------------------------------------------------------
__global__ __launch_bounds__(256) void deg_init_kernel(int* __restrict__ deg) {
    int i = blockIdx.x * 256 + threadIdx.x;
    if (i < N_NODES) deg[i] = 1;
}

__global__ __launch_bounds__(256) void deg_count_kernel(const long long* __restrict__ ei,
                                                        int* __restrict__ deg) {
    int e = blockIdx.x * 256 + threadIdx.x;
    if (e < N_EDGES) {
        long long d = ei[(long long)N_EDGES + e];
        atomicAdd(&deg[d], 1);
    }
}

__global__ __launch_bounds__(256) void dinv_kernel(const int* __restrict__ deg,
                                                   float* __restrict__ dinv) {
    int i = blockIdx.x * 256 + threadIdx.x;
    if (i < N_NODES) dinv[i] = rsqrtf((float)deg[i]);   // deg >= 1 always
}

__global__ __launch_bounds__(256) void zero_kernel(float* __restrict__ p, int n) {
    int i = blockIdx.x * 256 + threadIdx.x;
    if (i < n) p[i] = 0.0f;
}

// ---------------------------------------------------------------------------
// Fused dual GEMM: h_mu = x @ W_mu, h_lv = x @ W_logvar  (fp32 WMMA 16x16x4)
// Block = 8 waves. Waves 0..3 -> W_mu n-tiles 0..3, waves 4..7 -> W_logvar.
// x tile (16 rows x 256) staged in LDS with row stride 260 (bank-conflict-free
// A-frag reads, rows stay 16B aligned for b128 stores).
// ---------------------------------------------------------------------------
__global__ __launch_bounds__(256) void gemm_dual_wmma_kernel(
    const float* __restrict__ x,
    const float* __restrict__ Wmu,
    const float* __restrict__ Wlv,
    float* __restrict__ h_mu,
    float* __restrict__ h_lv) {

    __shared__ __align__(16) float lds_x[16 * 260];

    const int t = threadIdx.x;
    const long long row_base = (long long)blockIdx.x * 16;
    const float* xin = x + row_base * D_IN;

    // Cooperative load: 4096 floats as 1024 float4 (coalesced), re-strided to 260.
    #pragma unroll
    for (int i = 0; i < 4; ++i) {
        int idx = (t + i * 256) * 4;          // flat float index in [0,4096)
        int row = idx >> 8;                    // /256
        int col = idx & 255;
        v4f v = *(const v4f*)(xin + idx);
        *(v4f*)(&lds_x[row * 260 + col]) = v;
    }
    __syncthreads();

    const int wave = t >> 5;
    const int lane = t & 31;
    const float* W   = (wave < 4) ? Wmu  : Wlv;
    float*       hout = (wave < 4) ? h_mu : h_lv;
    const int n_base = (wave & 3) * 16;

    const int mrow = lane & 15;    // A: M = lane%16
    const int hi   = lane >> 4;    // A/B: hi half-wave takes K+2
    const int ncol = lane & 15;    // B/C: N = lane%16

    v8f c = {};
    const float* arow = &lds_x[mrow * 260];
    const float* bcol = W + n_base + ncol;

    for (int k = 0; k < D_IN; k += 4) {
        // A 16x4 f32 frag: VGPR v holds K = k + 2*hi + v
        v2f a, b;
        a.x = arow[k + 2 * hi];
        a.y = arow[k + 2 * hi + 1];
        // B 4x16 f32 frag (mirrored layout): VGPR v holds K = k + 2*hi + v, N = ncol
        b.x = bcol[(long long)(k + 2 * hi) * D_OUT];
        b.y = bcol[(long long)(k + 2 * hi + 1) * D_OUT];
        c = __builtin_amdgcn_wmma_f32_16x16x4_f32(false, a, false, b,
                                                  (short)0, c, false, false);
    }

    // C/D layout: VGPR r -> M = r + 8*hi, N = ncol
    #pragma unroll
    for (int r = 0; r < 8; ++r) {
        long long m = row_base + r + 8 * hi;
        hout[m * D_OUT + n_base + ncol] = c[r];
    }
}

// ---------------------------------------------------------------------------
// Edge scatter: one wave32 per edge; lane handles features {lane, lane+32}
// for both convs. Accumulates into mu/log_var sections of d_out via f32
// atomics (resolve in L2; all hot buffers fit the 192MB L2).
// ---------------------------------------------------------------------------
__global__ __launch_bounds__(256) void edge_scatter_kernel(
    const long long* __restrict__ ei,
    const float* __restrict__ dinv,
    const float* __restrict__ h_mu,
    const float* __restrict__ h_lv,
    float* __restrict__ acc_mu,
    float* __restrict__ acc_lv) {

    long long gtid = (long long)blockIdx.x * 256 + threadIdx.x;
    long long e = gtid >> 5;
    if (e >= N_EDGES) return;
    int lane = threadIdx.x & 31;

    long long s = ei[e];
    long long d = ei[(long long)N_EDGES + e];
    float norm = dinv[s] * dinv[d];

    const float* hm = h_mu + s * D_OUT;
    const float* hl = h_lv + s * D_OUT;
    float* am = acc_mu + d * D_OUT;
    float* al = acc_lv + d * D_OUT;

    atomicAdd(&am[lane],      norm * hm[lane]);
    atomicAdd(&am[lane + 32], norm * hm[lane + 32]);
    atomicAdd(&al[lane],      norm * hl[lane]);
    atomicAdd(&al[lane + 32], norm * hl[lane + 32]);
}

// ---------------------------------------------------------------------------
// Finalize: add self-loop term (norm = dinv^2), then z = mu + exp(0.5*lv)*eps.
// Runs after edge_scatter on the same stream -> non-atomic RMW is safe.
// ---------------------------------------------------------------------------
__global__ __launch_bounds__(256) void finalize_kernel(
    const float* __restrict__ dinv,
    const float* __restrict__ h_mu,
    const float* __restrict__ h_lv,
    const float* __restrict__ eps,
    float* __restrict__ z,
    float* __restrict__ mu_out,
    float* __restrict__ lv_out) {

    long long i = (long long)blockIdx.x * 256 + threadIdx.x;
    if (i >= (long long)N_NODES * D_OUT) return;
    long long node = i >> 6;   // /64
    float di = dinv[node];
    float ns = di * di;
    float mu = mu_out[i] + ns * h_mu[i];
    float lv = lv_out[i] + ns * h_lv[i];
    mu_out[i] = mu;
    lv_out[i] = lv;
    z[i] = mu + expf(0.5f * lv) * eps[i];
}

// ---------------------------------------------------------------------------
extern "C" void kernel_launch(void* const* d_in, const int* in_sizes, int n_in,
                              void* d_out, int out_size, void* d_ws, size_t ws_size,
                              hipStream_t stream) {
    const float*     x   = (const float*)d_in[0];
    const long long* ei  = (const long long*)d_in[1];   // int64 edge_index [2, E]
    const float*     eps = (const float*)d_in[2];
    const float*     Wmu = (const float*)d_in[3];
    const float*     Wlv = (const float*)d_in[4];

    const size_t NF = (size_t)N_NODES * D_OUT;          // 6,400,000
    float* z  = (float*)d_out;
    float* mu = z + NF;
    float* lv = mu + NF;

    char* ws = (char*)d_ws;
    int*   deg  = (int*)ws;                                    // 400,000 B
    float* dinv = (float*)(ws + (512 * 1024));                 // 400,000 B
    float* h_mu = (float*)(ws + (1024 * 1024));                // 25.6 MB
    float* h_lv = (float*)(ws + (1024 * 1024) + NF * sizeof(float));

    // 1) degrees + dinv (independent of GEMM)
    deg_init_kernel <<<(N_NODES + 255) / 256, 256, 0, stream>>>(deg);
    deg_count_kernel<<<(N_EDGES + 255) / 256, 256, 0, stream>>>(ei, deg);
    dinv_kernel     <<<(N_NODES + 255) / 256, 256, 0, stream>>>(deg, dinv);

    // 2) zero the mu/log_var accumulators (contiguous in d_out)
    zero_kernel<<<(int)((2 * NF + 255) / 256), 256, 0, stream>>>(mu, (int)(2 * NF));

    // 3) fused dual GEMM via fp32 WMMA (100000/16 = 6250 full tiles)
    gemm_dual_wmma_kernel<<<N_NODES / 16, 256, 0, stream>>>(x, Wmu, Wlv, h_mu, h_lv);

    // 4) edge gather/scale/scatter-add (wave per edge)
    long long threads = (long long)N_EDGES * 32;
    edge_scatter_kernel<<<(int)((threads + 255) / 256), 256, 0, stream>>>(
        ei, dinv, h_mu, h_lv, mu, lv);

    // 5) self-loop term + reparametrization
    finalize_kernel<<<(int)((NF + 255) / 256), 256, 0, stream>>>(
        dinv, h_mu, h_lv, eps, z, mu, lv);
}